// SupGCN_4011499454822
// MI455X (gfx1250) — compile-verified
//
#include <hip/hip_runtime.h>
#include <hip/hip_bf16.h>

// -------- sizes from the reference --------
#define N_NODES   50000
#define N_EDGES   800000
#define FDIM      64
#define N_GRAPHS  64
#define N_CLASSES 10

typedef __attribute__((ext_vector_type(2))) float v2f;
typedef __attribute__((ext_vector_type(8))) float v8f;

// ---------------------------------------------------------------------------
// zero fill
// ---------------------------------------------------------------------------
__global__ void zero_kernel(float* __restrict__ p, int n) {
    int i = blockIdx.x * blockDim.x + threadIdx.x;
    if (i < n) p[i] = 0.0f;
}

// ---------------------------------------------------------------------------
// Supernode pre-pass: x2[dst] += x[src]*edge_mask  ;  deg[dst] += 1 (unmasked)
// One thread per (edge, 4-feature quad): 16 threads/edge.
// ---------------------------------------------------------------------------
__global__ void scatter_x2_kernel(const float* __restrict__ x,
                                  const int* __restrict__ src,
                                  const int* __restrict__ dst,
                                  const float* __restrict__ emask,
                                  float* __restrict__ x2,
                                  float* __restrict__ deg) {
    int tid = blockIdx.x * blockDim.x + threadIdx.x;
    if (tid >= N_EDGES * 16) return;
    int e = tid >> 4;
    int q = (tid & 15) << 2;
    int s = src[e], d = dst[e];
    float m = emask[e];
    const float4 xv = *(const float4*)(x + (size_t)s * FDIM + q);
    float* o = x2 + (size_t)d * FDIM + q;
    atomicAdd(o + 0, xv.x * m);
    atomicAdd(o + 1, xv.y * m);
    atomicAdd(o + 2, xv.z * m);
    atomicAdd(o + 3, xv.w * m);
    if ((tid & 15) == 0) atomicAdd(deg + d, 1.0f);
}

// ---------------------------------------------------------------------------
// xbuf = supernode ? x2 : x   ;   dinv = rsqrt(deg + 1)
// ---------------------------------------------------------------------------
__global__ void select_deg_kernel(const float* __restrict__ x,
                                  const float* __restrict__ x2,
                                  const unsigned char* __restrict__ smask,
                                  const float* __restrict__ deg,
                                  float* __restrict__ xbuf,
                                  float* __restrict__ dinv) {
    int idx = blockIdx.x * blockDim.x + threadIdx.x;
    if (idx >= N_NODES * FDIM) return;
    int i = idx >> 6;
    xbuf[idx] = smask[i] ? x2[idx] : x[idx];
    if ((idx & 63) == 0) dinv[i] = rsqrtf(deg[i] + 1.0f);
}

// ---------------------------------------------------------------------------
// WMMA GEMM: H[N_NODES x 64] = X[N_NODES x 64] @ W[64 x 64]   (f32, exact)
// One wave per 16x16 output tile; 4 waves per block = 4 column tiles;
// grid.x = N_NODES/16 = 3125 row tiles. K=64 walked in steps of 4 via
// V_WMMA_F32_16X16X4_F32. EXEC is all-1s (no divergence, exact tiling).
//
// Operand layouts (ISA 7.12.2, 32-bit):
//   A 16x4 : lane L (half=L>>4, r=L&15): a[j] = A[r][k0 + 2*half + j]
//   B 4x16 : b[j] = B[k0 + 2*half + j][n0 + r]
//   C 16x16: acc[v] = D[v + 8*half][n0 + r]
// ---------------------------------------------------------------------------
__global__ void gemm64_wmma_kernel(const float* __restrict__ X,
                                   const float* __restrict__ W,
                                   float* __restrict__ H) {
    const int lane = threadIdx.x & 31;
    const int wave = threadIdx.x >> 5;        // 0..3 -> column tile
    const int half = lane >> 4;
    const int r    = lane & 15;
    const int m0   = blockIdx.x * 16;
    const int n0   = wave * 16;

    const float* __restrict__ xrow = X + (size_t)(m0 + r) * FDIM;

    v8f acc = {};
#pragma unroll
    for (int k0 = 0; k0 < FDIM; k0 += 4) {
        const int ka = k0 + half * 2;
        v2f a, b;
        a.x = xrow[ka];
        a.y = xrow[ka + 1];
        b.x = W[ka * FDIM + n0 + r];
        b.y = W[(ka + 1) * FDIM + n0 + r];
        acc = __builtin_amdgcn_wmma_f32_16x16x4_f32(
            /*neg_a=*/false, a, /*neg_b=*/false, b,
            /*c_mod=*/(short)0, acc, /*reuse_a=*/false, /*reuse_b=*/false);
    }
#pragma unroll
    for (int v = 0; v < 8; ++v) {
        H[(size_t)(m0 + v + half * 8) * FDIM + n0 + r] = acc[v];
    }
}

// ---------------------------------------------------------------------------
// agg[dst] += hlin[src] * (dinv[src]*dinv[dst]) per edge
// ---------------------------------------------------------------------------
__global__ void scatter_agg_kernel(const float* __restrict__ hlin,
                                   const int* __restrict__ src,
                                   const int* __restrict__ dst,
                                   const float* __restrict__ dinv,
                                   float* __restrict__ agg) {
    int tid = blockIdx.x * blockDim.x + threadIdx.x;
    if (tid >= N_EDGES * 16) return;
    int e = tid >> 4;
    int q = (tid & 15) << 2;
    int s = src[e], d = dst[e];
    float c = dinv[s] * dinv[d];
    const float4 hv = *(const float4*)(hlin + (size_t)s * FDIM + q);
    float* o = agg + (size_t)d * FDIM + q;
    atomicAdd(o + 0, hv.x * c);
    atomicAdd(o + 1, hv.y * c);
    atomicAdd(o + 2, hv.z * c);
    atomicAdd(o + 3, hv.w * c);
}

// ---------------------------------------------------------------------------
// xbuf = [relu]( agg + hlin * dinv^2 + b )
// ---------------------------------------------------------------------------
__global__ void layer_finish_kernel(const float* __restrict__ agg,
                                    const float* __restrict__ hlin,
                                    const float* __restrict__ dinv,
                                    const float* __restrict__ bias,
                                    float* __restrict__ xbuf,
                                    int do_relu) {
    int idx = blockIdx.x * blockDim.x + threadIdx.x;
    if (idx >= N_NODES * FDIM) return;
    int i = idx >> 6;
    int f = idx & 63;
    float di = dinv[i];
    float v = agg[idx] + hlin[idx] * (di * di) + bias[f];
    xbuf[idx] = do_relu ? fmaxf(v, 0.0f) : v;
}

// ---------------------------------------------------------------------------
// Mean-pool accumulate: psum[batch[i]] += h[i] ; cnt[batch[i]] += 1
// ---------------------------------------------------------------------------
__global__ void pool_kernel(const float* __restrict__ h,
                            const int* __restrict__ batch,
                            float* __restrict__ psum,
                            float* __restrict__ pcnt) {
    int idx = blockIdx.x * blockDim.x + threadIdx.x;
    if (idx >= N_NODES * FDIM) return;
    int i = idx >> 6;
    int f = idx & 63;
    int g = batch[i];
    atomicAdd(&psum[g * FDIM + f], h[idx]);
    if (f == 0) atomicAdd(&pcnt[g], 1.0f);
}

// ---------------------------------------------------------------------------
// out[g,c] = (psum[g]/max(cnt,1)) @ Wl + bl   (64x10, tiny)
// ---------------------------------------------------------------------------
__global__ void final_kernel(const float* __restrict__ psum,
                             const float* __restrict__ pcnt,
                             const float* __restrict__ Wl,
                             const float* __restrict__ bl,
                             float* __restrict__ out) {
    int tid = blockIdx.x * blockDim.x + threadIdx.x;
    if (tid >= N_GRAPHS * N_CLASSES) return;
    int g = tid / N_CLASSES, c = tid % N_CLASSES;
    float inv = 1.0f / fmaxf(pcnt[g], 1.0f);
    float acc = bl[c];
#pragma unroll
    for (int k = 0; k < FDIM; ++k)
        acc += psum[g * FDIM + k] * inv * Wl[k * N_CLASSES + c];
    out[tid] = acc;
}

// ---------------------------------------------------------------------------
// launch
// ---------------------------------------------------------------------------
extern "C" void kernel_launch(void* const* d_in, const int* in_sizes, int n_in,
                              void* d_out, int out_size, void* d_ws, size_t ws_size,
                              hipStream_t stream) {
    // Inputs (setup_inputs order)
    const float*         x     = (const float*)d_in[0];
    const int*           ei    = (const int*)d_in[1];           // [2, E]
    const unsigned char* smask = (const unsigned char*)d_in[2]; // jnp bool -> 1 byte
    const float*         emask = (const float*)d_in[3];
    const int*           batch = (const int*)d_in[4];
    const float*         W1    = (const float*)d_in[5];
    const float*         b1    = (const float*)d_in[6];
    const float*         W2    = (const float*)d_in[7];
    const float*         b2    = (const float*)d_in[8];
    const float*         W3    = (const float*)d_in[9];
    const float*         b3    = (const float*)d_in[10];
    const float*         Wl    = (const float*)d_in[11];
    const float*         bl    = (const float*)d_in[12];
    float* out = (float*)d_out;

    const int* src = ei;
    const int* dst = ei + N_EDGES;

    // Workspace layout (floats)
    float* ws   = (float*)d_ws;
    const size_t NF = (size_t)N_NODES * FDIM;        // 3.2M
    float* xbuf = ws;                                 // [N,64]
    float* hlin = xbuf + NF;                          // [N,64]
    float* agg  = hlin + NF;                          // [N,64] (also x2 scratch)
    float* deg  = agg + NF;                           // [N]
    float* dinv = deg + N_NODES;                      // [N]
    float* psum = dinv + N_NODES;                     // [64,64]
    float* pcnt = psum + N_GRAPHS * FDIM;             // [64]

    const int TB = 256;
    const int nNF   = (int)NF;
    const int gNF   = (nNF + TB - 1) / TB;
    const int nE16  = N_EDGES * 16;
    const int gE16  = (nE16 + TB - 1) / TB;

    // --- zero scratch used with atomics (every call: ws is not re-poisoned) ---
    zero_kernel<<<gNF, TB, 0, stream>>>(agg, nNF);                 // x2 scratch
    zero_kernel<<<(N_NODES + TB - 1) / TB, TB, 0, stream>>>(deg, N_NODES);
    zero_kernel<<<(N_GRAPHS * FDIM + N_GRAPHS + TB - 1) / TB, TB, 0, stream>>>(
        psum, N_GRAPHS * FDIM + N_GRAPHS);                         // psum + pcnt

    // --- supernode aggregation + degree ---
    scatter_x2_kernel<<<gE16, TB, 0, stream>>>(x, src, dst, emask, agg, deg);
    select_deg_kernel<<<gNF, TB, 0, stream>>>(x, agg, smask, deg, xbuf, dinv);

    // --- three GCN layers ---
    const float* Ws[3] = {W1, W2, W3};
    const float* bs[3] = {b1, b2, b3};
    for (int l = 0; l < 3; ++l) {
        gemm64_wmma_kernel<<<N_NODES / 16, 128, 0, stream>>>(xbuf, Ws[l], hlin);
        zero_kernel<<<gNF, TB, 0, stream>>>(agg, nNF);
        scatter_agg_kernel<<<gE16, TB, 0, stream>>>(hlin, src, dst, dinv, agg);
        layer_finish_kernel<<<gNF, TB, 0, stream>>>(agg, hlin, dinv, bs[l], xbuf,
                                                    (l < 2) ? 1 : 0);
    }

    // --- global mean pool + classifier ---
    pool_kernel<<<gNF, TB, 0, stream>>>(xbuf, batch, psum, pcnt);
    final_kernel<<<(N_GRAPHS * N_CLASSES + TB - 1) / TB, TB, 0, stream>>>(
        psum, pcnt, Wl, bl, out);
}